// STGCN_53403623358850
// MI455X (gfx1250) — compile-verified
//
#include <hip/hip_runtime.h>

#define T_   10
#define N_   10000
#define E_   320000
#define CH   32
#define NT   (N_ * T_)
#define NT32 (NT * CH)

typedef __attribute__((ext_vector_type(2))) float v2f;
typedef __attribute__((ext_vector_type(8))) float v8f;

__device__ __forceinline__ void atomAddF(float* p, float v) {
    unsafeAtomicAdd(p, v);  // native global_atomic_add_f32 on gfx1250
}

// ---------------- graph normalization ----------------

__global__ void k_initgraph(float* __restrict__ deg, float* __restrict__ expsum) {
    int i = blockIdx.x * blockDim.x + threadIdx.x;
    if (i < N_) deg[i] = 1.0f;          // self-loop weight
    if (i == 0) expsum[0] = 0.0f;
}

__global__ void k_expsum(const float* __restrict__ ew, float* __restrict__ expsum) {
    __shared__ float sb[256];
    int tid = threadIdx.x;
    int i = blockIdx.x * 256 + tid;
    sb[tid] = (i < E_) ? __expf(ew[i]) : 0.0f;
    __syncthreads();
    for (int s = 128; s > 0; s >>= 1) {
        if (tid < s) sb[tid] += sb[tid + s];
        __syncthreads();
    }
    if (tid == 0) atomAddF(expsum, sb[0]);
}

__global__ void k_deg(const int* __restrict__ ei, const float* __restrict__ ew,
                      const float* __restrict__ expsum, float* __restrict__ deg) {
    int e = blockIdx.x * 256 + threadIdx.x;
    if (e >= E_) return;
    float inv = 1.0f / expsum[0];
    atomAddF(&deg[ei[E_ + e]], __expf(ew[e]) * inv);
}

__global__ void k_dinv(const float* __restrict__ deg, float* __restrict__ dinv,
                       float* __restrict__ nself) {
    int i = blockIdx.x * 256 + threadIdx.x;
    if (i >= N_) return;
    float d = deg[i];                    // >= 1 always (self loop)
    dinv[i] = rsqrtf(d);
    nself[i] = 1.0f / d;                 // dinv^2 for the self-loop edge
}

__global__ void k_norm(const int* __restrict__ ei, const float* __restrict__ ew,
                       const float* __restrict__ expsum, const float* __restrict__ dinv,
                       float* __restrict__ nrm) {
    int e = blockIdx.x * 256 + threadIdx.x;
    if (e >= E_) return;
    float inv = 1.0f / expsum[0];
    nrm[e] = dinv[ei[e]] * (__expf(ew[e]) * inv) * dinv[ei[E_ + e]];
}

// ---------------- stage A1: temporal conv + instance norm + relu ----------------
// input x[t*st + n*sn + ci]; output H node-major [(n*T+t)*32 + co]
__global__ void k_stageA1(const float* __restrict__ X, long sn, long st, int Cin,
                          const float* __restrict__ tw, const float* __restrict__ tb,
                          const float* __restrict__ gamma, const float* __restrict__ beta,
                          float* __restrict__ H) {
    __shared__ float xs[8 * T_ * CH];        // [nl][t][ci], Cin <= 32
    __shared__ float wsl[CH * 3 * CH];       // [(ci*3+k)*32 + co]  (bank-conflict-free)
    const int tid = threadIdx.x;
    const int co = tid & 31;
    const int nl = tid >> 5;                 // 0..7 local node
    const int node0 = blockIdx.x * 8;

    for (int idx = tid; idx < CH * Cin * 3; idx += 256) {
        int k = idx % 3;
        int ci = (idx / 3) % Cin;
        int c = idx / (3 * Cin);
        wsl[(ci * 3 + k) * 32 + c] = tw[idx];
    }
    int tot = 8 * T_ * Cin;
    for (int idx = tid; idx < tot; idx += 256) {
        int ci = idx % Cin;
        int t = (idx / Cin) % T_;
        int n = idx / (Cin * T_);
        xs[(n * T_ + t) * Cin + ci] = X[(long)t * st + (long)(node0 + n) * sn + ci];
    }
    __syncthreads();

    const float* xn = xs + nl * T_ * Cin;
    float y[T_];
    float bias = tb[co];
    #pragma unroll
    for (int t = 0; t < T_; ++t) {
        float acc = bias;
        #pragma unroll
        for (int k = 0; k < 3; ++k) {
            int tt = t + k - 1;
            if (tt < 0 || tt >= T_) continue;
            const float* xr = xn + tt * Cin;
            const float* wr = wsl + k * 32 + co;
            for (int ci = 0; ci < Cin; ++ci)
                acc += wr[ci * 96] * xr[ci];
        }
        y[t] = fminf(fmaxf(acc, -10.0f), 10.0f);
    }
    float mu = 0.0f;
    #pragma unroll
    for (int t = 0; t < T_; ++t) mu += y[t];
    mu *= 0.1f;
    float var = 0.0f;
    #pragma unroll
    for (int t = 0; t < T_; ++t) { float d = y[t] - mu; var += d * d; }
    var *= 0.1f;
    float sc = gamma[co] * rsqrtf(var + 1e-5f);
    float bb = beta[co];
    long base = ((long)(node0 + nl) * T_) * CH + co;
    #pragma unroll
    for (int t = 0; t < T_; ++t)
        H[base + (long)t * CH] = fmaxf((y[t] - mu) * sc + bb, 0.0f);
}

// ---------------- WMMA GEMM: D[NT,32] = A[NT,32] * B[32,32] (f32, exact) ----------------
__global__ void k_gemm32(const float* __restrict__ A, const float* __restrict__ B,
                         float* __restrict__ D, int ntiles) {
    int lane = threadIdx.x & 31;
    int wv = blockIdx.x * (blockDim.x >> 5) + (threadIdx.x >> 5);
    if (wv >= ntiles) return;                 // wave-uniform: EXEC stays all-ones
    int hf = lane >> 4;                       // half: 0 = lanes 0-15, 1 = lanes 16-31
    int lr = lane & 15;
    long rbase = (long)wv * 16;
    v8f c0 = {}, c1 = {};
    const float* arow = A + (rbase + lr) * CH;
    #pragma unroll
    for (int k = 0; k < 8; ++k) {
        int kk = 4 * k + 2 * hf;
        v2f a;  a.x  = arow[kk];              a.y  = arow[kk + 1];
        v2f b0; b0.x = B[kk * 32 + lr];       b0.y = B[(kk + 1) * 32 + lr];
        v2f b1; b1.x = B[kk * 32 + 16 + lr];  b1.y = B[(kk + 1) * 32 + 16 + lr];
        c0 = __builtin_amdgcn_wmma_f32_16x16x4_f32(false, a, false, b0, (short)0, c0, false, false);
        c1 = __builtin_amdgcn_wmma_f32_16x16x4_f32(false, a, false, b1, (short)0, c1, false, false);
    }
    #pragma unroll
    for (int i = 0; i < 8; ++i) {
        long r = rbase + i + 8 * hf;
        D[r * CH + lr]      = c0[i];
        D[r * CH + 16 + lr] = c1[i];
    }
}

// ---------------- WMMA projection: D[NT,16] = A[NT,32] * B[32,16] ----------------
__global__ void k_proj(const float* __restrict__ A, const float* __restrict__ B,
                       float* __restrict__ D, int ntiles) {
    int lane = threadIdx.x & 31;
    int wv = blockIdx.x * (blockDim.x >> 5) + (threadIdx.x >> 5);
    if (wv >= ntiles) return;
    int hf = lane >> 4;
    int lr = lane & 15;
    long rbase = (long)wv * 16;
    v8f c0 = {};
    const float* arow = A + (rbase + lr) * CH;
    #pragma unroll
    for (int k = 0; k < 8; ++k) {
        int kk = 4 * k + 2 * hf;
        v2f a;  a.x  = arow[kk];         a.y  = arow[kk + 1];
        v2f b0; b0.x = B[kk * 16 + lr];  b0.y = B[(kk + 1) * 16 + lr];
        c0 = __builtin_amdgcn_wmma_f32_16x16x4_f32(false, a, false, b0, (short)0, c0, false, false);
    }
    #pragma unroll
    for (int i = 0; i < 8; ++i) {
        long r = rbase + i + 8 * hf;
        D[r * 16 + lr] = c0[i];
    }
}

// ---------------- zero buffer ----------------
__global__ void k_zero(float* __restrict__ p, int len) {
    int i = blockIdx.x * 256 + threadIdx.x;
    if (i < len) p[i] = 0.0f;
}

// ---------------- edge scatter: one wave32 per edge, lane = channel ----------------
__global__ void k_scatter(const int* __restrict__ ei, const float* __restrict__ nrm,
                          const float* __restrict__ nself, const float* __restrict__ HW,
                          float* __restrict__ AG) {
    int lane = threadIdx.x & 31;
    long wv = (long)blockIdx.x * (blockDim.x >> 5) + (threadIdx.x >> 5);
    long nw = (long)gridDim.x * (blockDim.x >> 5);
    for (long e = wv; e < (long)(E_ + N_); e += nw) {
        int r, c;
        float w;
        if (e < E_) { r = ei[e]; c = ei[E_ + e]; w = nrm[e]; }
        else        { r = c = (int)(e - E_);     w = nself[e - E_]; }
        const float* src = HW + ((long)r * T_) * CH + lane;   // coalesced 128B per t
        float* dst = AG + ((long)c * T_) * CH + lane;
        #pragma unroll
        for (int t = 0; t < T_; ++t)
            atomAddF(dst + t * CH, src[t * CH] * w);
    }
}

// ---------------- stage C: + gb, relu, clip; node-major out ----------------
__global__ void k_stageC(const float* __restrict__ AG, const float* __restrict__ gb,
                         float* __restrict__ OUT) {
    int i = blockIdx.x * 256 + threadIdx.x;
    if (i >= NT32) return;
    float v = AG[i] + gb[i & 31];
    OUT[i] = fminf(fmaxf(v, 0.0f), 10.0f);
}

// ---------------- mean over T + bias ----------------
__global__ void k_mean(const float* __restrict__ TMP, const float* __restrict__ ob,
                       float* __restrict__ OUT) {
    int i = blockIdx.x * 256 + threadIdx.x;
    if (i >= N_ * 16) return;
    int n = i >> 4, d = i & 15;
    float acc = 0.0f;
    #pragma unroll
    for (int t = 0; t < T_; ++t)
        acc += TMP[((long)n * T_ + t) * 16 + d];
    OUT[i] = acc * 0.1f + ob[d];
}

extern "C" void kernel_launch(void* const* d_in, const int* in_sizes, int n_in,
                              void* d_out, int out_size, void* d_ws, size_t ws_size,
                              hipStream_t stream) {
    const float* x        = (const float*)d_in[0];
    const int*   ei       = (const int*)d_in[1];
    const float* ew       = (const float*)d_in[2];
    const float* l1_tw    = (const float*)d_in[3];
    const float* l1_tb    = (const float*)d_in[4];
    const float* l1_gw    = (const float*)d_in[5];
    const float* l1_gb    = (const float*)d_in[6];
    const float* l1_gamma = (const float*)d_in[7];
    const float* l1_beta  = (const float*)d_in[8];
    const float* l2_tw    = (const float*)d_in[9];
    const float* l2_tb    = (const float*)d_in[10];
    const float* l2_gw    = (const float*)d_in[11];
    const float* l2_gb    = (const float*)d_in[12];
    const float* l2_gamma = (const float*)d_in[13];
    const float* l2_beta  = (const float*)d_in[14];
    const float* out_w    = (const float*)d_in[15];
    const float* out_b    = (const float*)d_in[16];
    float* out = (float*)d_out;

    float* B0     = (float*)d_ws;
    float* B1     = B0 + NT32;
    float* B2     = B1 + NT32;
    float* nrm    = B2 + NT32;
    float* nself  = nrm + E_;
    float* deg    = nself + N_;
    float* dinv   = deg + N_;
    float* expsum = dinv + N_;

    const int ntiles = NT / 16;                  // 6250
    const int gB = (ntiles + 7) / 8;             // 8 waves / block

    // graph normalization (shared by both layers)
    k_initgraph<<<(N_ + 255) / 256, 256, 0, stream>>>(deg, expsum);
    k_expsum<<<(E_ + 255) / 256, 256, 0, stream>>>(ew, expsum);
    k_deg<<<(E_ + 255) / 256, 256, 0, stream>>>(ei, ew, expsum, deg);
    k_dinv<<<(N_ + 255) / 256, 256, 0, stream>>>(deg, dinv, nself);
    k_norm<<<(E_ + 255) / 256, 256, 0, stream>>>(ei, ew, expsum, dinv, nrm);

    // layer 1 (Cin = 16, input layout [T, N, 16])
    k_stageA1<<<N_ / 8, 256, 0, stream>>>(x, 16L, (long)N_ * 16, 16,
                                          l1_tw, l1_tb, l1_gamma, l1_beta, B0);
    k_gemm32<<<gB, 256, 0, stream>>>(B0, l1_gw, B1, ntiles);
    k_zero<<<(NT32 + 255) / 256, 256, 0, stream>>>(B2, NT32);
    k_scatter<<<2048, 256, 0, stream>>>(ei, nrm, nself, B1, B2);
    k_stageC<<<(NT32 + 255) / 256, 256, 0, stream>>>(B2, l1_gb, B0);

    // layer 2 (Cin = 32, input node-major [n][t][32])
    k_stageA1<<<N_ / 8, 256, 0, stream>>>(B0, (long)(T_ * CH), (long)CH, CH,
                                          l2_tw, l2_tb, l2_gamma, l2_beta, B1);
    k_gemm32<<<gB, 256, 0, stream>>>(B1, l2_gw, B2, ntiles);
    k_zero<<<(NT32 + 255) / 256, 256, 0, stream>>>(B0, NT32);
    k_scatter<<<2048, 256, 0, stream>>>(ei, nrm, nself, B2, B0);
    k_stageC<<<(NT32 + 255) / 256, 256, 0, stream>>>(B0, l2_gb, B1);

    // output projection (WMMA) + temporal mean
    k_proj<<<gB, 256, 0, stream>>>(B1, out_w, B2, ntiles);
    k_mean<<<(N_ * 16 + 255) / 256, 256, 0, stream>>>(B2, out_b, out);
}